// HouseholderPerm_5652176962001
// MI455X (gfx1250) — compile-verified
//
#include <hip/hip_runtime.h>

typedef __attribute__((ext_vector_type(2))) float v2f;
typedef __attribute__((ext_vector_type(8))) float v8f;

#define WIDTH  512
#define NREFL  32
#define VPITCH 33   // padded LDS pitch (33 coprime w/ 64 banks -> conflict-free)
#define GPITCH 33

// One block per batch sample. 256 threads = 8 wave32.
// Compact-WY reformulation of 32 sequential Householder reflections:
//   G = V^T V (32x32 Gram, via v_wmma_f32_16x16x4_f32)
//   p = x V
//   alpha_i = (2/G_ii) * (p_i - sum_{j<i} G_ji alpha_j)
//   out = x - V alpha
__global__ __launch_bounds__(256)
void householder_wy_kernel(const float* __restrict__ x,
                           const float* __restrict__ c,
                           float* __restrict__ out)
{
    __shared__ __align__(16) float sV[WIDTH * VPITCH];  // reflectors, padded rows
    __shared__ __align__(16) float sX[WIDTH];           // input row
    __shared__ float sG[NREFL * GPITCH];                // Gram matrix
    __shared__ float sPpart[128];                       // partial p = x.V
    __shared__ float sAlpha[NREFL];

    const int tid  = threadIdx.x;
    const int b    = blockIdx.x;
    const int lane = tid & 31;
    const int wave = tid >> 5;

    // ---- Stage V (= c row, [512 x 32] row-major) into padded LDS; read c ONCE ----
    const float* crow = c + (size_t)b * (WIDTH * NREFL);
    #pragma unroll
    for (int it = 0; it < 16; ++it) {
        int g = it * 1024 + tid * 4;               // 256 thr * 4 floats = 1024/iter
        float4 v4 = *reinterpret_cast<const float4*>(crow + g);
        int row = g >> 5, col = g & 31;            // col % 4 == 0, no row crossing
        float* dst = &sV[row * VPITCH + col];
        dst[0] = v4.x; dst[1] = v4.y; dst[2] = v4.z; dst[3] = v4.w;
    }
    if (tid < 128) {
        float4 v4 = *reinterpret_cast<const float4*>(x + (size_t)b * WIDTH + tid * 4);
        *reinterpret_cast<float4*>(&sX[tid * 4]) = v4;
    }
    __syncthreads();

    if (wave < 4) {
        // ---- Gram tile (16x16) per wave: G[16ti.., 16tj..] = sum_k V^T V ----
        const int ti = wave >> 1, tj = wave & 1;
        const int half = lane >> 4, lm = lane & 15;
        const int ca = ti * 16 + lm;   // A-matrix column (row of G tile)
        const int cb = tj * 16 + lm;   // B-matrix column (col of G tile)
        v8f acc = {};
        for (int kk = 0; kk < WIDTH / 4; ++kk) {
            // f32 16x4 A layout: lanes 0-15 hold K=0(vgpr0),K=1(vgpr1);
            //                    lanes 16-31 hold K=2,K=3. B mirrors (4x16).
            int r0 = kk * 4 + half * 2;
            v2f a, bb;
            a.x  = sV[(r0    ) * VPITCH + ca];
            a.y  = sV[(r0 + 1) * VPITCH + ca];
            bb.x = sV[(r0    ) * VPITCH + cb];
            bb.y = sV[(r0 + 1) * VPITCH + cb];
            acc = __builtin_amdgcn_wmma_f32_16x16x4_f32(
                      false, a, false, bb, (short)0, acc, false, false);
        }
        // C/D layout: VGPR r -> row M=r (lanes 0-15), M=r+8 (lanes 16-31)
        #pragma unroll
        for (int r = 0; r < 8; ++r) {
            int row = ti * 16 + half * 8 + r;
            sG[row * GPITCH + cb] = acc[r];
        }
    } else {
        // ---- p = x . V : 4 waves, thread (seg, r) sums 128 rows ----
        int t2  = tid - 128;
        int r   = t2 & 31;
        int seg = t2 >> 5;
        float s = 0.0f;
        int w0 = seg * 128;
        #pragma unroll 4
        for (int w = w0; w < w0 + 128; ++w)
            s += sX[w] * sV[w * VPITCH + r];
        sPpart[t2] = s;
    }
    __syncthreads();

    // ---- Triangular alpha recurrence on wave 0 (32 steps, wave32 reductions) ----
    if (tid < 32) {
        const int j = lane;
        float pj = sPpart[j] + sPpart[32 + j] + sPpart[64 + j] + sPpart[96 + j];
        float aj = 0.0f;
        for (int i = 0; i < NREFL; ++i) {
            float contrib = (j < i) ? sG[j * GPITCH + i] * aj : 0.0f;
            #pragma unroll
            for (int m = 16; m >= 1; m >>= 1)
                contrib += __shfl_xor(contrib, m, 32);
            if (j == i) {
                aj = (2.0f / sG[i * GPITCH + i]) * (pj - contrib);
                sAlpha[i] = aj;
            }
        }
    }
    __syncthreads();

    // ---- out = x - V * alpha ----
    float* orow = out + (size_t)b * WIDTH;
    #pragma unroll
    for (int h = 0; h < 2; ++h) {
        int w = tid + h * 256;
        float acc = sX[w];
        #pragma unroll
        for (int j = 0; j < NREFL; ++j)
            acc -= sAlpha[j] * sV[w * VPITCH + j];
        orow[w] = acc;
    }
}

extern "C" void kernel_launch(void* const* d_in, const int* in_sizes, int n_in,
                              void* d_out, int out_size, void* d_ws, size_t ws_size,
                              hipStream_t stream) {
    const float* x = (const float*)d_in[0];
    const float* c = (const float*)d_in[1];
    float* out = (float*)d_out;
    const int batch = in_sizes[0] / WIDTH;   // 16384
    householder_wy_kernel<<<dim3(batch), dim3(256), 0, stream>>>(x, c, out);
}